// GNNEncoder_46067819217044
// MI455X (gfx1250) — compile-verified
//
#include <hip/hip_runtime.h>

#define N_NODES 50000
#define N_EDGES 600000
#define N_PAD   50048   /* 782 * 64, zero-padded rows keep WMMA unpredicated */
#define D       128
#define EPS     1e-5f

typedef __attribute__((ext_vector_type(2))) float v2f;
typedef __attribute__((ext_vector_type(8))) float v8f;

// ---- one-time per call: E_agg[n,0:6] = sum of edgeFeats into dst, deg[n] = in-degree ----
__global__ void edge_prep_kernel(const long long* __restrict__ ei,
                                 const float* __restrict__ ef,
                                 float* __restrict__ Eagg,
                                 float* __restrict__ deg) {
  int e = blockIdx.x * 256 + threadIdx.x;
  if (e >= N_EDGES) return;
  int dst = (int)ei[2 * e + 1];
  const float* f = ef + (size_t)e * 6;
  float* row = Eagg + (size_t)dst * 8;
#pragma unroll
  for (int k = 0; k < 6; ++k) atomicAdd(&row[k], f[k]);
  atomicAdd(&deg[dst], 1.0f);
}

// ---- Y[dst, :din] += x[src, :din]   (one wave32 per edge) ----
__global__ void scatter_kernel(const float* __restrict__ x, int ldx, int din,
                               const long long* __restrict__ ei,
                               float* __restrict__ Y) {
  int e = blockIdx.x * 8 + (threadIdx.x >> 5);
  if (e >= N_EDGES) return;
  int lane = threadIdx.x & 31;
  int src = (int)ei[2 * e + 0];
  int dst = (int)ei[2 * e + 1];
  const float* xr = x + (size_t)src * ldx;
  float* yr = Y + (size_t)dst * D;
  for (int k = lane; k < din; k += 32)
    atomicAdd(&yr[k], xr[k]);
}

// ---- H = relu(Y@W + Eagg@We + deg (x) (b+be)), f32 WMMA 16x16x4 ----
// block = 128 threads (4 waves); each wave computes a 16x128 strip.
__global__ void __launch_bounds__(128) gemm_relu_kernel(
    const float* __restrict__ Y,      // N_PAD x 128 (cols >= K are zero)
    const float* __restrict__ Wm,     // K x 128
    const float* __restrict__ Eagg,   // N_PAD x 8 (cols 6,7 zero)
    const float* __restrict__ Wem,    // 6 x 128
    const float* __restrict__ bn,     // 128
    const float* __restrict__ bev,    // 128
    const float* __restrict__ deg,    // N_PAD
    float* __restrict__ H,            // N_PAD x 128
    int K, int KP) {
  extern __shared__ float smem[];
  float* sW  = smem;            // 64 x 128 chunk of W
  float* sWe = smem + 64 * D;   // 8 x 128 (zero-padded rows 6,7)
  const int tid = threadIdx.x;

  for (int i = tid; i < 8 * D; i += 128) {
    int r = i >> 7;
    sWe[i] = (r < 6) ? Wem[i] : 0.f;
  }

  const int lane = tid & 31;
  const int wv   = tid >> 5;
  const int row0 = blockIdx.x * 64 + wv * 16;
  const int m    = lane & 15;          // M (A) / N (B,C,D) within tile
  const int kg   = (lane >> 4) << 1;   // K group: lanes 16-31 hold K+2,K+3
  const int hi   = (lane >> 4) << 3;   // C/D row offset: lanes 16-31 hold M+8

  v8f acc[8];
  v8f z = {};
#pragma unroll
  for (int t = 0; t < 8; ++t) acc[t] = z;

  const float* Ar = Y + (size_t)(row0 + m) * D;

  for (int kc = 0; kc < KP; kc += 64) {
    int kcn = KP - kc; if (kcn > 64) kcn = 64;
    __syncthreads();
    for (int i = tid; i < kcn * D; i += 128) {
      int rl = i >> 7, c = i & 127;
      int rg = kc + rl;
      sW[i] = (rg < K) ? Wm[(size_t)rg * D + c] : 0.f;
    }
    __syncthreads();
    for (int k = 0; k < kcn; k += 4) {
      v2f a;
      a.x = Ar[kc + k + kg];
      a.y = Ar[kc + k + kg + 1];
#pragma unroll
      for (int t = 0; t < 8; ++t) {
        v2f b;
        b.x = sW[(k + kg) * D + t * 16 + m];
        b.y = sW[(k + kg + 1) * D + t * 16 + m];
        acc[t] = __builtin_amdgcn_wmma_f32_16x16x4_f32(
            false, a, false, b, (short)0, acc[t], false, false);
      }
    }
  }

  // edge-aggregate term, K padded 6 -> 8
  const float* Er = Eagg + (size_t)(row0 + m) * 8;
#pragma unroll
  for (int k = 0; k < 8; k += 4) {
    v2f a;
    a.x = Er[k + kg];
    a.y = Er[k + kg + 1];
#pragma unroll
    for (int t = 0; t < 8; ++t) {
      v2f b;
      b.x = sWe[(k + kg) * D + t * 16 + m];
      b.y = sWe[(k + kg + 1) * D + t * 16 + m];
      acc[t] = __builtin_amdgcn_wmma_f32_16x16x4_f32(
          false, a, false, b, (short)0, acc[t], false, false);
    }
  }

  float dv[8];
#pragma unroll
  for (int v = 0; v < 8; ++v) dv[v] = deg[row0 + hi + v];

#pragma unroll
  for (int t = 0; t < 8; ++t) {
    int col = t * 16 + m;
    float bias = bn[col] + bev[col];
#pragma unroll
    for (int v = 0; v < 8; ++v) {
      float val = acc[t][v] + dv[v] * bias;
      H[(size_t)(row0 + hi + v) * D + col] = fmaxf(val, 0.f);
    }
  }
}

// ---- per-column partial sums / sum-of-squares over real rows ----
__global__ void colstats_kernel(const float* __restrict__ H,
                                float* __restrict__ colsum,
                                float* __restrict__ colsq) {
  int col = threadIdx.x & 127;
  int row = blockIdx.x * 2 + (threadIdx.x >> 7);
  int stride = gridDim.x * 2;
  float s = 0.f, q = 0.f;
  for (int r = row; r < N_NODES; r += stride) {
    float v = H[(size_t)r * D + col];
    s += v; q += v * v;
  }
  atomicAdd(&colsum[col], s);
  atomicAdd(&colsq[col], q);
}

__global__ void finalize_kernel(const float* __restrict__ colsum,
                                const float* __restrict__ colsq,
                                const float* __restrict__ g,
                                const float* __restrict__ bb,
                                float* __restrict__ scale,
                                float* __restrict__ shift) {
  int c = threadIdx.x;
  if (c >= D) return;
  float mean = colsum[c] * (1.0f / N_NODES);
  float var  = colsq[c] * (1.0f / N_NODES) - mean * mean;
  if (var < 0.f) var = 0.f;
  float inv = rsqrtf(var + EPS);
  float sc = g[c] * inv;
  scale[c] = sc;
  shift[c] = bb[c] - mean * sc;
}

__global__ void normalize_kernel(float* __restrict__ H,
                                 const float* __restrict__ scale,
                                 const float* __restrict__ shift) {
  size_t i = (size_t)blockIdx.x * 256 + threadIdx.x;
  int c = (int)(i & 127);
  H[i] = H[i] * scale[c] + shift[c];
}

__global__ void out_kernel(const float* __restrict__ colsum, float* __restrict__ out) {
  int c = threadIdx.x;
  if (c >= D) return;
  out[c] = colsum[c] * (1.0f / N_NODES);
}

extern "C" void kernel_launch(void* const* d_in, const int* in_sizes, int n_in,
                              void* d_out, int out_size, void* d_ws, size_t ws_size,
                              hipStream_t stream) {
  const float*     nodeF = (const float*)d_in[0];
  const long long* eidx  = (const long long*)d_in[1];   // int64 per reference
  const float*     ef    = (const float*)d_in[2];
  const float* W[3]  = {(const float*)d_in[3],  (const float*)d_in[9],  (const float*)d_in[15]};
  const float* b[3]  = {(const float*)d_in[4],  (const float*)d_in[10], (const float*)d_in[16]};
  const float* We[3] = {(const float*)d_in[5],  (const float*)d_in[11], (const float*)d_in[17]};
  const float* be[3] = {(const float*)d_in[6],  (const float*)d_in[12], (const float*)d_in[18]};
  const float* g[3]  = {(const float*)d_in[7],  (const float*)d_in[13], (const float*)d_in[19]};
  const float* bb[3] = {(const float*)d_in[8],  (const float*)d_in[14], (const float*)d_in[20]};

  float* ws   = (float*)d_ws;
  float* xA   = ws;                                 // N_PAD * 128
  float* xB   = xA   + (size_t)N_PAD * D;           // N_PAD * 128
  float* Yb   = xB   + (size_t)N_PAD * D;           // N_PAD * 128
  float* Eagg = Yb   + (size_t)N_PAD * D;           // N_PAD * 8
  float* deg  = Eagg + (size_t)N_PAD * 8;           // N_PAD
  float* colsum = deg + N_PAD;                      // 128
  float* colsq  = colsum + D;                       // 128
  float* scaleb = colsq + D;                        // 128
  float* shiftb = scaleb + D;                       // 128

  // layer-invariant edge aggregation
  hipMemsetAsync(Eagg, 0, ((size_t)N_PAD * 8 + N_PAD) * sizeof(float), stream);
  edge_prep_kernel<<<(N_EDGES + 255) / 256, 256, 0, stream>>>(eidx, ef, Eagg, deg);

  float* xcur = nullptr;
  float* xnxt = xA;
  const int smemBytes = (64 * D + 8 * D) * (int)sizeof(float);  // 36 KB

  for (int l = 0; l < 3; ++l) {
    int K   = (l == 0) ? 19 : D;
    int KP  = (l == 0) ? 20 : D;      // K rounded up to multiple of 4
    int ldx = (l == 0) ? 19 : D;
    const float* xin = (l == 0) ? nodeF : xcur;

    hipMemsetAsync(Yb, 0, (size_t)N_PAD * D * sizeof(float), stream);
    scatter_kernel<<<(N_EDGES + 7) / 8, 256, 0, stream>>>(xin, ldx, K, eidx, Yb);
    gemm_relu_kernel<<<N_PAD / 64, 128, smemBytes, stream>>>(
        Yb, W[l], Eagg, We[l], b[l], be[l], deg, xnxt, K, KP);

    hipMemsetAsync(colsum, 0, 2 * D * sizeof(float), stream);
    colstats_kernel<<<256, 256, 0, stream>>>(xnxt, colsum, colsq);
    finalize_kernel<<<1, D, 0, stream>>>(colsum, colsq, g[l], bb[l], scaleb, shiftb);
    normalize_kernel<<<(N_PAD * D) / 256, 256, 0, stream>>>(xnxt, scaleb, shiftb);

    xcur = xnxt;
    xnxt = (xnxt == xA) ? xB : xA;
  }

  // output = column mean of final normalized features
  hipMemsetAsync(colsum, 0, 2 * D * sizeof(float), stream);
  colstats_kernel<<<256, 256, 0, stream>>>(xcur, colsum, colsq);
  out_kernel<<<1, D, 0, stream>>>(colsum, (float*)d_out);
}